// ECELoss_66855460930055
// MI455X (gfx1250) — compile-verified
//
#include <hip/hip_runtime.h>
#include <stdint.h>
#include <math.h>

#define NBINS 15
#define NC 100                        // classes per row
#define TILE_ROWS 32                  // one row per lane (wave32)
#define TILE_FLOATS (TILE_ROWS * NC)  // 3200 floats
#define TILE_BYTES  (TILE_FLOATS * 4) // 12800 B
#define NB128 25                      // 25 x b128 async copies = 12800 B exactly
#define WAVES_PER_BLOCK 2
#define BLOCK_THREADS (WAVES_PER_BLOCK * 32)
#define VEC_LANES 25                  // tail path: 25 lanes * float4 = 100 floats

// ---------------------------------------------------------------------------
// Async global->LDS tile copy (GV mode, 64-bit vaddr, saddr = off).
// the offset adds to both the LDS and the global address -> one immediate
// offset serves both sides. 25 x b128 (32 lanes * 16B = 512B each) move one
// 12800 B tile. Tracked on ASYNCcnt.
// ---------------------------------------------------------------------------
template <int I>
__device__ __forceinline__ void issue_b128(uint32_t la, uint64_t ga) {
    if constexpr (I < NB128) {
        asm volatile("global_load_async_to_lds_b128 %0, %1, off offset:%2"
                     :: "v"(la), "v"(ga), "i"(I * 512)
                     : "memory");
        issue_b128<I + 1>(la, ga);
    }
}

__device__ __forceinline__ void issue_tile_loads(uint64_t gbase, uint32_t lbase, int lane) {
    issue_b128<0>(lbase + (uint32_t)(lane * 16), gbase + (uint64_t)(lane * 16));
}

__device__ __forceinline__ void accumulate_row(float m, int idx, int label, float* sbins) {
    float conf = 1.0f / (1.0f + expf(-m));          // sigmoid(max logit)
    float acc  = (idx == label) ? 1.0f : 0.0f;
    int b = (int)ceilf(conf * (float)NBINS) - 1;    // searchsorted('left') - 1
    b = min(max(b, 0), NBINS - 1);
    atomicAdd(&sbins[b], 1.0f);                     // count
    atomicAdd(&sbins[NBINS + b], conf);             // conf sum
    atomicAdd(&sbins[2 * NBINS + b], acc);          // acc sum
}

// Transposed processing: lane l owns row l of the 32-row tile and serially
// scans its 100 floats from LDS. No cross-lane reduction, all lanes active,
// strict '>' preserves first-occurrence argmax semantics.
__device__ __forceinline__ void process_tile(const float* __restrict__ buf, int lane,
                                             const int* __restrict__ labels,
                                             long long row0, float* sbins) {
    const float4* p = reinterpret_cast<const float4*>(buf + lane * NC);
    float m = -1.0e30f;
    int idx = 0;
#pragma unroll
    for (int i = 0; i < NC / 4; ++i) {
        float4 v = p[i];
        int base = i * 4;
        if (v.x > m) { m = v.x; idx = base; }
        if (v.y > m) { m = v.y; idx = base + 1; }
        if (v.z > m) { m = v.z; idx = base + 2; }
        if (v.w > m) { m = v.w; idx = base + 3; }
    }
    accumulate_row(m, idx, labels[row0 + lane], sbins);  // coalesced label load
}

// Wave-wide max + first-index argmax (wave32 butterfly) -- tail path only.
__device__ __forceinline__ void wave_argmax(float& m, int& idx) {
#pragma unroll
    for (int off = 16; off >= 1; off >>= 1) {
        float om = __shfl_xor(m, off, 32);
        int   oi = __shfl_xor(idx, off, 32);
        if (om > m || (om == m && oi < idx)) { m = om; idx = oi; }
    }
}

// ---------------------------------------------------------------------------
// Main kernel: async double-buffered streaming of logit tiles through LDS.
// ---------------------------------------------------------------------------
__global__ void __launch_bounds__(BLOCK_THREADS)
ece_main_kernel(const float* __restrict__ logits, const int* __restrict__ labels,
                float* __restrict__ gbins, long long N) {
    __shared__ alignas(16) float sbuf[WAVES_PER_BLOCK][2][TILE_FLOATS]; // 51200 B
    __shared__ float sbins[3 * NBINS];

    const int tid  = threadIdx.x;
    const int lane = tid & 31;
    const int w    = tid >> 5;

    if (tid < 3 * NBINS) sbins[tid] = 0.0f;
    __syncthreads();

    const long long tiles = N / TILE_ROWS;
    const long long wid = (long long)blockIdx.x * WAVES_PER_BLOCK + w;
    const long long nw  = (long long)gridDim.x * WAVES_PER_BLOCK;

    const uint32_t lbase0 = (uint32_t)(uintptr_t)&sbuf[w][0][0];
    const uint32_t lbase1 = (uint32_t)(uintptr_t)&sbuf[w][1][0];
    const uint64_t gB = (uint64_t)(uintptr_t)logits;

    int cur = 0;
    if (wid < tiles) {
        issue_tile_loads(gB + (uint64_t)wid * TILE_BYTES, lbase0, lane);
    }
    for (long long t = wid; t < tiles; t += nw) {
        const long long tn = t + nw;
        if (tn < tiles) {
            // prefetch next tile into the other buffer, then wait for current
            issue_tile_loads(gB + (uint64_t)tn * TILE_BYTES, cur ? lbase0 : lbase1, lane);
            asm volatile("s_wait_asynccnt 25" ::: "memory");
        } else {
            asm volatile("s_wait_asynccnt 0" ::: "memory");
        }
        process_tile(&sbuf[w][cur][0], lane, labels, t * TILE_ROWS, sbins);
        cur ^= 1;
    }

    // Tail rows (N % 32 != 0): direct coalesced global float4 loads, one wave
    // per row with a butterfly reduction.
    const long long tail0 = tiles * TILE_ROWS;
    const long long rem = N - tail0;
    if (wid < rem) {
        const long long row = tail0 + wid;
        float m = -1.0e30f;
        int idx = 0x7FFFFFFF;
        if (lane < VEC_LANES) {
            const float4* p = reinterpret_cast<const float4*>(logits + row * NC);
            float4 v = p[lane];
            int base = lane * 4;
            m = v.x; idx = base;
            if (v.y > m) { m = v.y; idx = base + 1; }
            if (v.z > m) { m = v.z; idx = base + 2; }
            if (v.w > m) { m = v.w; idx = base + 3; }
        }
        wave_argmax(m, idx);
        if (lane == 0) {
            accumulate_row(m, idx, labels[row], sbins);
        }
    }

    __syncthreads();
    if (tid < 3 * NBINS) {
        atomicAdd(&gbins[tid], sbins[tid]);
    }
}

__global__ void ece_init_kernel(float* __restrict__ gbins) {
    if (threadIdx.x < 3 * NBINS) gbins[threadIdx.x] = 0.0f;
}

__global__ void ece_finalize_kernel(const float* __restrict__ gbins,
                                    float* __restrict__ out, long long N) {
    if (threadIdx.x == 0 && blockIdx.x == 0) {
        const float n = (float)N;
        float ece = 0.0f;
        for (int b = 0; b < NBINS; ++b) {
            float c = gbins[b];
            if (c > 0.0f) {
                float sc = fmaxf(c, 1.0f);
                float avg_conf = gbins[NBINS + b] / sc;
                float avg_acc  = gbins[2 * NBINS + b] / sc;
                ece += (avg_conf - avg_acc) * (c / n);
            }
        }
        out[0] = ece;
    }
}

extern "C" void kernel_launch(void* const* d_in, const int* in_sizes, int n_in,
                              void* d_out, int out_size, void* d_ws, size_t ws_size,
                              hipStream_t stream) {
    const float* logits = (const float*)d_in[0];
    const int*   labels = (const int*)d_in[1];
    float* gbins = (float*)d_ws;     // 45 floats of scratch
    float* out   = (float*)d_out;

    const long long N = (long long)in_sizes[1];   // rows == label count

    ece_init_kernel<<<1, 64, 0, stream>>>(gbins);
    ece_main_kernel<<<4096, BLOCK_THREADS, 0, stream>>>(logits, labels, gbins, N);
    ece_finalize_kernel<<<1, 1, 0, stream>>>(gbins, out, N);
}